// LieRNN_84232898609665
// MI455X (gfx1250) — compile-verified
//
#include <hip/hip_runtime.h>
#include <hip/hip_bf16.h>

// LieRNN on MI455X (gfx1250), full fp32 via V_WMMA_F32_16X16X4_F32.
// B=16, T=4096, REC1=256, HID=16.
//
// Pipeline:
//   1) encode:      h[bt] = relu(x[bt,:] @ Wenc_flat)     (WMMA GEMM, TDM-staged A)
//   2) scan_local:  per (b, chunk of 64): in-place prefix products (WMMA)
//   3) scan_agg:    per b: exclusive scan of 64 chunk totals       (WMMA)
//   4) apply_decode: hiddens = E @ local; out = relu(hiddens @ WdecT) (WMMA)
//
// 64-row blocks amortize weight-matrix (B operand) reads 4x to keep L2
// traffic ~256 MB per GEMM, versus a ~6 us HBM roofline for the 128 MB of
// mandatory x/out traffic.

#define BATCH 16
#define TLEN  4096
#define REC   256
#define HID   16
#define MTOT  (BATCH * TLEN)   // 65536 rows of the big GEMMs
#define CHUNK 64
#define NCHUNK (TLEN / CHUNK)  // 64 chunks per batch

typedef __attribute__((ext_vector_type(2))) float v2f;
typedef __attribute__((ext_vector_type(8))) float v8f;
typedef __attribute__((ext_vector_type(4))) unsigned int v4u;
typedef __attribute__((ext_vector_type(8))) int v8i;
typedef __attribute__((ext_vector_type(4))) int v4i;

#if defined(__has_builtin)
#  if __has_builtin(__builtin_amdgcn_tensor_load_to_lds)
#    define USE_TDM 1
#  endif
#endif
#ifndef USE_TDM
#  define USE_TDM 0
#endif

// D = A(16x4,f32) * B(4x16,f32) + C(16x16,f32)
__device__ __forceinline__ v8f wmma4(v2f a, v2f b, v8f c) {
  return __builtin_amdgcn_wmma_f32_16x16x4_f32(
      /*neg_a=*/false, a, /*neg_b=*/false, b,
      /*c_mod=*/(short)0, c, /*reuse_a=*/false, /*reuse_b=*/false);
}

__device__ __forceinline__ float relu(float v) { return v > 0.f ? v : 0.f; }

// ---------------------------------------------------------------------------
// Kernel 1: encode GEMM. grid = MTOT/64 row blocks, 512 threads (16 waves).
// The 64x256 f32 x-tile (64 KB) is staged into LDS by the Tensor Data Mover.
// Wave w computes column tile w for all 4 row tiles (B slice reused 4x).
// ---------------------------------------------------------------------------
__global__ __launch_bounds__(512) void encode_kernel(
    const float* __restrict__ x, const float* __restrict__ wenc,
    float* __restrict__ h) {
  __shared__ __align__(16) float xt[64 * REC];  // 64 rows x 256 K = 64 KB
  const int tid  = threadIdx.x;
  const int lane = tid & 31;
  const int wave = tid >> 5;
  const long rowbase = (long)blockIdx.x * 64;

#if USE_TDM
  if (wave == 0) {
    // Build a 2D tensor descriptor: tile = 256 (dim0) x 64 (dim1) f32,
    // tensor_dim0_stride = 256 elements, destination = xt.
    const unsigned long long ga =
        (unsigned long long)(const void*)(x + rowbase * REC);
    const unsigned ldsa = (unsigned)(uintptr_t)(void*)xt;
    v4u g0;
    g0.x = 1u;                                   // count=1, user descriptor
    g0.y = ldsa;                                 // lds_addr
    g0.z = (unsigned)ga;                         // global_addr[31:0]
    g0.w = (unsigned)((ga >> 32) & 0x01FFFFFFull) | 0x80000000u;  // type=2
    v8i g1;
    g1[0] = 0x00020000;          // workgroup_mask=0, data_size=2 (4 bytes)
    g1[1] = (int)(256u << 16);   // tensor_dim0 = 256 (bits 63:48)
    g1[2] = (int)(64u << 16);    // tensor_dim1 = 64  (bits 95:80)
    g1[3] = (int)(256u << 16);   // tile_dim0 = 256   (bits 127:112)
    g1[4] = 64;                  // tile_dim1 = 64
    g1[5] = 256;                 // tensor_dim0_stride = 256
    g1[6] = 0;
    g1[7] = 0;
    v4i z4 = {0, 0, 0, 0};
    v8i z8 = {0, 0, 0, 0, 0, 0, 0, 0};
    // 6-arg toolchain variant: (g0, g1, g2, g3, g4, cpol)
    __builtin_amdgcn_tensor_load_to_lds(g0, g1, z4, z4, z8, 0);
    __builtin_amdgcn_s_wait_tensorcnt(0);
  }
#else
  {
    const float4* xg = (const float4*)(x + rowbase * REC);
    float4* xs = (float4*)xt;
#pragma unroll
    for (int i = 0; i < 8; ++i) xs[tid + 512 * i] = xg[tid + 512 * i];
  }
#endif
  __syncthreads();

  const int colbase = wave * 16;
  const int row  = lane & 15;        // M index (A), N index (B)
  const int koff = (lane >> 4) * 2;  // K-pair select per half-wave

  v8f acc0 = {}, acc1 = {}, acc2 = {}, acc3 = {};
  for (int k0 = 0; k0 < REC; k0 += 4) {
    v2f b;
    b.x = wenc[(long)(k0 + koff) * REC + colbase + row];
    b.y = wenc[(long)(k0 + koff + 1) * REC + colbase + row];
    v2f a0 = *(const v2f*)&xt[(row +  0) * REC + k0 + koff];
    v2f a1 = *(const v2f*)&xt[(row + 16) * REC + k0 + koff];
    v2f a2 = *(const v2f*)&xt[(row + 32) * REC + k0 + koff];
    v2f a3 = *(const v2f*)&xt[(row + 48) * REC + k0 + koff];
    acc0 = wmma4(a0, b, acc0);
    acc1 = wmma4(a1, b, acc1);
    acc2 = wmma4(a2, b, acc2);
    acc3 = wmma4(a3, b, acc3);
  }

  const int mhalf = (lane >> 4) * 8;
#pragma unroll
  for (int r = 0; r < 8; ++r) {
    const long col = colbase + row;
    h[(rowbase +  0 + r + mhalf) * (long)REC + col] = relu(acc0[r]);
    h[(rowbase + 16 + r + mhalf) * (long)REC + col] = relu(acc1[r]);
    h[(rowbase + 32 + r + mhalf) * (long)REC + col] = relu(acc2[r]);
    h[(rowbase + 48 + r + mhalf) * (long)REC + col] = relu(acc3[r]);
  }
}

// ---------------------------------------------------------------------------
// Kernel 2: per-chunk local prefix products, in place over h.
// grid = BATCH*NCHUNK blocks of one wave (32 threads).
// ---------------------------------------------------------------------------
__global__ __launch_bounds__(32) void scan_local_kernel(
    float* __restrict__ h, float* __restrict__ agg) {
  __shared__ __align__(16) float P[16 * 16];  // running product, row-major
  const int lane = threadIdx.x;
  const int b = blockIdx.x >> 6;
  const int c = blockIdx.x & (NCHUNK - 1);
  const long base = ((long)b * TLEN + (long)c * CHUNK) * (HID * HID);

  const int row   = lane & 15;
  const int koff  = (lane >> 4) * 2;
  const int mhalf = (lane >> 4) * 8;

  // P = h[t0]  (coalesced 256-float copy to LDS)
  {
    const float4* src = (const float4*)(h + base);
    float4* dst = (float4*)P;
    dst[lane]      = src[lane];
    dst[lane + 32] = src[lane + 32];
  }
  __syncthreads();

  for (int t = 1; t < CHUNK; ++t) {
    const float* ht = h + base + (long)t * 256;
    if (t + 1 < CHUNK)
      __builtin_prefetch(h + base + (long)(t + 1) * 256, 0, 3);
    v8f acc = {};
#pragma unroll
    for (int kk = 0; kk < 4; ++kk) {
      const int k0 = kk * 4;
      v2f a = *(const v2f*)&P[row * 16 + k0 + koff];  // P is A operand
      v2f bv;
      bv.x = ht[(k0 + koff) * 16 + row];              // h[t] is B operand
      bv.y = ht[(k0 + koff + 1) * 16 + row];
      acc = wmma4(a, bv, acc);
    }
    __syncthreads();  // all lanes done reading old P
    float* outp = h + base + (long)t * 256;
#pragma unroll
    for (int r = 0; r < 8; ++r) {
      const int m = r + mhalf;
      P[m * 16 + row]    = acc[r];  // new running product
      outp[m * 16 + row] = acc[r];  // local prefix, stored in place
    }
    __syncthreads();
  }

  // chunk aggregate = final P
  {
    float4* dst = (float4*)(agg + ((long)b * NCHUNK + c) * 256);
    const float4* src = (const float4*)P;
    dst[lane]      = src[lane];
    dst[lane + 32] = src[lane + 32];
  }
}

// ---------------------------------------------------------------------------
// Kernel 3: exclusive scan of chunk aggregates per batch. grid = BATCH waves.
// pref[b][c] = agg[b][0] @ ... @ agg[b][c-1], pref[b][0] = I.
// ---------------------------------------------------------------------------
__global__ __launch_bounds__(32) void scan_agg_kernel(
    const float* __restrict__ agg, float* __restrict__ pref) {
  __shared__ __align__(16) float P[16 * 16];
  const int lane = threadIdx.x;
  const int b = blockIdx.x;
  const int row   = lane & 15;
  const int koff  = (lane >> 4) * 2;
  const int mhalf = (lane >> 4) * 8;

  const float* aggb = agg + (long)b * NCHUNK * 256;
  float* prefb = pref + (long)b * NCHUNK * 256;

  // E[0] = identity
#pragma unroll
  for (int r = 0; r < 8; ++r) {
    const int m = r + mhalf;
    const float v = (m == row) ? 1.f : 0.f;
    P[m * 16 + row] = v;
    prefb[m * 16 + row] = v;
  }
  __syncthreads();

  for (int c = 1; c < NCHUNK; ++c) {
    const float* A = aggb + (long)(c - 1) * 256;
    v8f acc = {};
#pragma unroll
    for (int kk = 0; kk < 4; ++kk) {
      const int k0 = kk * 4;
      v2f a = *(const v2f*)&P[row * 16 + k0 + koff];
      v2f bv;
      bv.x = A[(k0 + koff) * 16 + row];
      bv.y = A[(k0 + koff + 1) * 16 + row];
      acc = wmma4(a, bv, acc);
    }
    __syncthreads();
    float* outp = prefb + (long)c * 256;
#pragma unroll
    for (int r = 0; r < 8; ++r) {
      const int m = r + mhalf;
      P[m * 16 + row]    = acc[r];
      outp[m * 16 + row] = acc[r];
    }
    __syncthreads();
  }
}

// ---------------------------------------------------------------------------
// Kernel 4: hiddens = E @ local, then out = relu(hiddens_flat @ Wdec^T).
// grid = MTOT/64 (one block per scan chunk!), 512 threads (16 waves).
// Phase 1: wave w applies the chunk's single E to local[t] for 4 t's
//          (E A-slices held in registers), filling a 64x256 decode A-tile.
// Phase 2: wave w does decode column tile w for all 4 row tiles, so every
//          Wdec B slice (contiguous b64) is reused by 4 WMMAs.
// ---------------------------------------------------------------------------
__global__ __launch_bounds__(512) void apply_decode_kernel(
    const float* __restrict__ local, const float* __restrict__ pref,
    const float* __restrict__ wdec, float* __restrict__ out) {
  __shared__ __align__(16) float At[64 * REC];  // 64 t-rows x 256 jk = 64 KB
  const int tid  = threadIdx.x;
  const int lane = tid & 31;
  const int wave = tid >> 5;
  const long t0 = (long)blockIdx.x * 64;  // flat (b*T + t) row base == chunk
  const int b = blockIdx.x >> 6;          // NCHUNK = 64 chunks per batch
  const int c = blockIdx.x & (NCHUNK - 1);
  const float* E = pref + ((long)b * NCHUNK + c) * 256;

  const int row   = lane & 15;
  const int koff  = (lane >> 4) * 2;
  const int mhalf = (lane >> 4) * 8;

  // E A-slices, loaded once, reused by all 4 apply matmuls of this wave.
  v2f ea0 = *(const v2f*)&E[row * 16 +  0 + koff];
  v2f ea1 = *(const v2f*)&E[row * 16 +  4 + koff];
  v2f ea2 = *(const v2f*)&E[row * 16 +  8 + koff];
  v2f ea3 = *(const v2f*)&E[row * 16 + 12 + koff];

#pragma unroll
  for (int u = 0; u < 4; ++u) {
    const int t = wave * 4 + u;
    const float* L = local + (t0 + t) * 256L;
    v8f acc = {};
    v2f bv;
    bv.x = L[(0 + koff) * 16 + row];  bv.y = L[(1 + koff) * 16 + row];
    acc = wmma4(ea0, bv, acc);
    bv.x = L[(4 + koff) * 16 + row];  bv.y = L[(5 + koff) * 16 + row];
    acc = wmma4(ea1, bv, acc);
    bv.x = L[(8 + koff) * 16 + row];  bv.y = L[(9 + koff) * 16 + row];
    acc = wmma4(ea2, bv, acc);
    bv.x = L[(12 + koff) * 16 + row]; bv.y = L[(13 + koff) * 16 + row];
    acc = wmma4(ea3, bv, acc);
#pragma unroll
    for (int r = 0; r < 8; ++r)
      At[t * REC + (r + mhalf) * 16 + row] = acc[r];  // flatten jk into K dim
  }
  __syncthreads();

  // Decode GEMM, column tile = wave. B[i][o] = wdec[o*256 + i] -> the
  // per-lane (K,K+1) pair is a contiguous b64 load from wdec.
  const int colbase = wave * 16;
  v8f acc0 = {}, acc1 = {}, acc2 = {}, acc3 = {};
  for (int k0 = 0; k0 < REC; k0 += 4) {
    v2f bv = *(const v2f*)&wdec[(long)(colbase + row) * REC + k0 + koff];
    v2f a0 = *(const v2f*)&At[(row +  0) * REC + k0 + koff];
    v2f a1 = *(const v2f*)&At[(row + 16) * REC + k0 + koff];
    v2f a2 = *(const v2f*)&At[(row + 32) * REC + k0 + koff];
    v2f a3 = *(const v2f*)&At[(row + 48) * REC + k0 + koff];
    acc0 = wmma4(a0, bv, acc0);
    acc1 = wmma4(a1, bv, acc1);
    acc2 = wmma4(a2, bv, acc2);
    acc3 = wmma4(a3, bv, acc3);
  }
#pragma unroll
  for (int r = 0; r < 8; ++r) {
    const long col = colbase + row;
    out[(t0 +  0 + r + mhalf) * (long)REC + col] = relu(acc0[r]);
    out[(t0 + 16 + r + mhalf) * (long)REC + col] = relu(acc1[r]);
    out[(t0 + 32 + r + mhalf) * (long)REC + col] = relu(acc2[r]);
    out[(t0 + 48 + r + mhalf) * (long)REC + col] = relu(acc3[r]);
  }
}

// ---------------------------------------------------------------------------
extern "C" void kernel_launch(void* const* d_in, const int* in_sizes, int n_in,
                              void* d_out, int out_size, void* d_ws, size_t ws_size,
                              hipStream_t stream) {
  (void)in_sizes; (void)n_in; (void)out_size; (void)ws_size;
  const float* x    = (const float*)d_in[0];  // (16,4096,256)
  const float* wenc = (const float*)d_in[1];  // (256,16,16)
  const float* wdec = (const float*)d_in[2];  // (256,16,16)
  float* out = (float*)d_out;                 // (16,4096,256)

  // Workspace layout: h/local (64 MB) | agg (1 MB) | pref (1 MB)
  float* h    = (float*)d_ws;
  float* agg  = h + (size_t)MTOT * 256;
  float* pref = agg + (size_t)BATCH * NCHUNK * 256;

  encode_kernel<<<MTOT / 64, 512, 0, stream>>>(x, wenc, h);
  scan_local_kernel<<<BATCH * NCHUNK, 32, 0, stream>>>(h, agg);
  scan_agg_kernel<<<BATCH, 32, 0, stream>>>(agg, pref);
  apply_decode_kernel<<<MTOT / 64, 512, 0, stream>>>(h, pref, wdec, out);
}